// MambaClassifier_11647951307048
// MI455X (gfx1250) — compile-verified
//
#include <hip/hip_runtime.h>
#include <hip/hip_bf16.h>
#include <cstddef>

// ---------------- problem constants ----------------
#define BB 64
#define TT 1000
#define CIN 12
#define DMODEL 128
#define DSTATE 16
#define DCONV 4
#define DINNER 256
#define DTRANK 8
#define NCLS 5
#define NROWS (BB * TT)       // 64000, multiple of 16
#define NPROJ (DTRANK + 2 * DSTATE)   // 40
#define NPROJ_PAD 48          // 3 x 16 col tiles
#define TTILES 63             // ceil(1000/16) time tiles per batch

typedef __attribute__((ext_vector_type(2))) float v2f;
typedef __attribute__((ext_vector_type(8))) float v8f;

__device__ __forceinline__ float siluf(float x) {
    return x / (1.0f + expf(-x));
}
__device__ __forceinline__ float softplusf(float x) {
    return (x > 20.0f) ? x : log1pf(expf(x));
}

// ---------------------------------------------------------------------------
// Kernel 1: fused ip (12->128) + in_proj first half (128->256) via WMMA f32.
// One block = 256 threads = 8 waves handles a 16-row tile of the 64000 rows.
// Each wave owns two 16x16 output tiles (8 waves x 2 = 256 cols).
// ---------------------------------------------------------------------------
__global__ __launch_bounds__(256) void k_inproj_wmma(
    const float* __restrict__ x,        // (B,T,12)
    const float* __restrict__ ip_w,     // (128,12)
    const float* __restrict__ ip_b,     // (128)
    const float* __restrict__ in_proj_w,// (512,128) row-major [n][k]
    float* __restrict__ xm)             // (B*T,256)
{
    __shared__ float hs[16][DMODEL + 4];   // 16 rows x 128 cols of h, padded

    const int tid     = threadIdx.x;
    const int rowbase = blockIdx.x * 16;

    // Stage 1: h = x @ ip_w.T + ip_b  (16 x 128 tile), 8 outputs/thread
    for (int i = tid; i < 16 * DMODEL; i += 256) {
        const int r = i >> 7;          // row in tile
        const int c = i & (DMODEL - 1);
        const float* xr = x + (size_t)(rowbase + r) * CIN;
        float acc = ip_b[c];
        #pragma unroll
        for (int k = 0; k < CIN; ++k)
            acc = fmaf(xr[k], ip_w[c * CIN + k], acc);
        hs[r][c] = acc;
    }
    __syncthreads();

    // Stage 2: xm_tile = h_tile @ in_proj_w[0:256].T  via V_WMMA_F32_16X16X4_F32
    const int wave    = tid >> 5;
    const int lane    = tid & 31;
    const int lane_lo = lane & 15;
    const int khalf   = (lane >> 4) * 2;   // lanes 16..31 carry K+2,K+3

    for (int ct = 0; ct < 2; ++ct) {
        const int n0 = (wave * 2 + ct) * 16;     // output column tile base
        v8f acc = {};
        #pragma unroll 4
        for (int k0 = 0; k0 < DMODEL; k0 += 4) {
            v2f a;
            a.x = hs[lane_lo][k0 + khalf + 0];
            a.y = hs[lane_lo][k0 + khalf + 1];
            const float* wp = in_proj_w + (size_t)(n0 + lane_lo) * DMODEL + k0 + khalf;
            v2f bf;
            bf.x = wp[0];
            bf.y = wp[1];
            acc = __builtin_amdgcn_wmma_f32_16x16x4_f32(
                false, a, false, bf, (short)0, acc, false, false);
        }
        const int mbase = (lane >> 4) * 8;
        #pragma unroll
        for (int v = 0; v < 8; ++v) {
            xm[(size_t)(rowbase + mbase + v) * DINNER + n0 + lane_lo] = acc[v];
        }
    }
}

// ---------------------------------------------------------------------------
// Kernel 2: one block per (b, 16-timestep tile).
//   stage A: depthwise causal conv(4)+bias+silu -> u tile (16x256) in LDS
//            (sliding 4-tap register window per channel)
//   stage B: dbc = u @ x_proj_w.T (16 x 40, padded to 48) via WMMA f32,
//            3 col-tiles on waves 0..2, K=256
//   stage C: dt_proj (8->256) + softplus -> delta; emit delta, du=delta*u,
//            B_t; at t=T-1 also C_last and u_last.
// ---------------------------------------------------------------------------
__global__ __launch_bounds__(256) void k_conv_proj_wmma(
    const float* __restrict__ xm,        // (B*T,256)
    const float* __restrict__ conv_w,    // (256,4)
    const float* __restrict__ conv_b,    // (256)
    const float* __restrict__ x_proj_w,  // (40,256)
    const float* __restrict__ dt_proj_w, // (256,8)
    const float* __restrict__ dt_proj_b, // (256)
    float* __restrict__ delta_o,         // (B*T,256)
    float* __restrict__ du_o,            // (B*T,256)
    float* __restrict__ Bm_o,            // (B*T,16)
    float* __restrict__ C_last,          // (B,16)
    float* __restrict__ u_last)          // (B,256)
{
    __shared__ float us[16][DINNER + 4];      // u tile, padded
    __shared__ float dbc[16][NPROJ_PAD];      // x_proj outputs

    const int tid  = threadIdx.x;
    const int b    = blockIdx.x / TTILES;
    const int tile = blockIdx.x % TTILES;
    const int t0   = tile * 16;

    // ---- stage A: conv + silu, sliding window; d = tid ----
    {
        const int d = tid;
        const float* base = xm + (size_t)b * TT * DINNER + d;
        const float w0 = conv_w[d * 4 + 0];
        const float w1 = conv_w[d * 4 + 1];
        const float w2 = conv_w[d * 4 + 2];
        const float w3 = conv_w[d * 4 + 3];
        const float cb = conv_b[d];
        float x3 = (t0 - 3 >= 0) ? base[(size_t)(t0 - 3) * DINNER] : 0.0f;
        float x2 = (t0 - 2 >= 0) ? base[(size_t)(t0 - 2) * DINNER] : 0.0f;
        float x1 = (t0 - 1 >= 0) ? base[(size_t)(t0 - 1) * DINNER] : 0.0f;
        #pragma unroll
        for (int r = 0; r < 16; ++r) {
            const int t = t0 + r;
            if (t < TT) {
                const float x0 = base[(size_t)t * DINNER];
                float acc = cb;
                acc = fmaf(x3, w0, acc);
                acc = fmaf(x2, w1, acc);
                acc = fmaf(x1, w2, acc);
                acc = fmaf(x0, w3, acc);
                us[r][d] = siluf(acc);
                x3 = x2; x2 = x1; x1 = x0;
            } else {
                us[r][d] = 0.0f;    // zero-pad rows past T
            }
        }
    }
    __syncthreads();

    // ---- stage B: dbc = u @ x_proj_w.T via WMMA (waves 0..2) ----
    const int wave    = tid >> 5;
    const int lane    = tid & 31;
    const int lane_lo = lane & 15;
    const int khalf   = (lane >> 4) * 2;

    if (wave < 3) {
        const int n0 = wave * 16;
        const int n  = n0 + lane_lo;
        const bool valid_n = (n < NPROJ);
        const float* wr = x_proj_w + (size_t)(valid_n ? n : 0) * DINNER;
        v8f acc = {};
        #pragma unroll 4
        for (int k0 = 0; k0 < DINNER; k0 += 4) {
            v2f a;
            a.x = us[lane_lo][k0 + khalf + 0];
            a.y = us[lane_lo][k0 + khalf + 1];
            v2f bf;
            bf.x = valid_n ? wr[k0 + khalf + 0] : 0.0f;
            bf.y = valid_n ? wr[k0 + khalf + 1] : 0.0f;
            acc = __builtin_amdgcn_wmma_f32_16x16x4_f32(
                false, a, false, bf, (short)0, acc, false, false);
        }
        const int mbase = (lane >> 4) * 8;
        #pragma unroll
        for (int v = 0; v < 8; ++v)
            dbc[mbase + v][n0 + lane_lo] = acc[v];
    }
    __syncthreads();

    // ---- stage C: dt_proj + softplus + outputs; d = tid ----
    {
        const int d = tid;
        const float bias = dt_proj_b[d];
        float wk[DTRANK];
        #pragma unroll
        for (int k = 0; k < DTRANK; ++k) wk[k] = dt_proj_w[d * DTRANK + k];

        #pragma unroll 4
        for (int r = 0; r < 16; ++r) {
            const int t = t0 + r;
            if (t >= TT) break;
            float dp = bias;
            #pragma unroll
            for (int k = 0; k < DTRANK; ++k)
                dp = fmaf(dbc[r][k], wk[k], dp);
            const float dl = softplusf(dp);
            const size_t row = (size_t)b * TT + t;
            const float u = us[r][d];
            delta_o[row * DINNER + d] = dl;
            du_o[row * DINNER + d]    = dl * u;
            if (t == TT - 1) u_last[b * DINNER + d] = u;
        }
    }
    // B_t rows: thread -> (r = tid>>4, n = tid&15)
    {
        const int r = tid >> 4;
        const int n = tid & 15;
        const int t = t0 + r;
        if (t < TT) {
            const size_t row = (size_t)b * TT + t;
            Bm_o[row * DSTATE + n] = dbc[r][DTRANK + n];
            if (t == TT - 1)
                C_last[b * DSTATE + n] = dbc[r][DTRANK + DSTATE + n];
        }
    }
}

// ---------------------------------------------------------------------------
// Kernel 3: z at t=T-1 only: recompute h row, z = h @ in_proj_w[256:512].T
// ---------------------------------------------------------------------------
__global__ __launch_bounds__(256) void k_zlast(
    const float* __restrict__ x,
    const float* __restrict__ ip_w,
    const float* __restrict__ ip_b,
    const float* __restrict__ in_proj_w,
    float* __restrict__ z_last)          // (B,256)
{
    __shared__ float hsl[DMODEL];
    const int b   = blockIdx.x;
    const int tid = threadIdx.x;
    if (tid < DMODEL) {
        const float* xr = x + ((size_t)b * TT + (TT - 1)) * CIN;
        float acc = ip_b[tid];
        #pragma unroll
        for (int k = 0; k < CIN; ++k)
            acc = fmaf(xr[k], ip_w[tid * CIN + k], acc);
        hsl[tid] = acc;
    }
    __syncthreads();
    const float* wr = in_proj_w + (size_t)(DINNER + tid) * DMODEL;
    float s = 0.0f;
    #pragma unroll 8
    for (int k = 0; k < DMODEL; ++k) s = fmaf(hsl[k], wr[k], s);
    z_last[b * DINNER + tid] = s;
}

// ---------------------------------------------------------------------------
// Kernel 4: selective scan. One lane per (b,d,n): 1000 serial steps of
// h = exp(delta*A)*h + (delta*u)*B_t. Final y only at t=T-1 via C_last,
// reduced over n with 16-lane xor-shuffles, fused with D-skip and silu(z).
// ---------------------------------------------------------------------------
__global__ __launch_bounds__(256) void k_scan(
    const float* __restrict__ delta_i,   // (B*T,256)
    const float* __restrict__ du_i,      // (B*T,256)
    const float* __restrict__ Bm_i,      // (B*T,16)
    const float* __restrict__ A_log,     // (256,16)
    const float* __restrict__ C_last,    // (B,16)
    const float* __restrict__ u_last,    // (B,256)
    const float* __restrict__ D_skip,    // (256)
    const float* __restrict__ z_last,    // (B,256)
    float* __restrict__ y_last)          // (B,256)
{
    const int i = blockIdx.x * 256 + threadIdx.x;   // 262144 lanes total
    const int n = i & (DSTATE - 1);
    const int d = (i >> 4) & (DINNER - 1);
    const int b = i >> 12;

    const float a = -expf(A_log[d * DSTATE + n]);   // constant over t

    const float* dptr = delta_i + (size_t)b * TT * DINNER + d;
    const float* uptr = du_i    + (size_t)b * TT * DINNER + d;
    const float* bptr = Bm_i    + (size_t)b * TT * DSTATE + n;

    float h = 0.0f;
    #pragma unroll 4
    for (int t = 0; t < TT; ++t) {
        const float dl = dptr[(size_t)t * DINNER];   // broadcast within n-group
        const float dA = expf(dl * a);
        const float in = uptr[(size_t)t * DINNER] * bptr[(size_t)t * DSTATE];
        h = fmaf(dA, h, in);
    }

    float v = h * C_last[b * DSTATE + n];
    #pragma unroll
    for (int m = 8; m >= 1; m >>= 1) v += __shfl_xor(v, m, 16);

    if (n == 0) {
        const float ul = u_last[b * DINNER + d];
        const float z  = z_last[b * DINNER + d];
        y_last[b * DINNER + d] = (v + ul * D_skip[d]) * siluf(z);
    }
}

// ---------------------------------------------------------------------------
// Kernel 5: classifier head on 64 rows: out_proj(256->128) -> fc1(->64,relu)
// -> fc2(->5). One block per batch.
// ---------------------------------------------------------------------------
__global__ __launch_bounds__(256) void k_head(
    const float* __restrict__ y_last,
    const float* __restrict__ out_proj_w, // (128,256)
    const float* __restrict__ fc1_w,      // (64,128)
    const float* __restrict__ fc1_b,
    const float* __restrict__ fc2_w,      // (5,64)
    const float* __restrict__ fc2_b,
    float* __restrict__ out)              // (B,5)
{
    __shared__ float ys[DINNER];
    __shared__ float os[DMODEL];
    __shared__ float hid[64];
    const int b = blockIdx.x;
    const int tid = threadIdx.x;

    ys[tid] = y_last[b * DINNER + tid];
    __syncthreads();
    if (tid < DMODEL) {
        const float* w = out_proj_w + (size_t)tid * DINNER;
        float s = 0.0f;
        #pragma unroll 8
        for (int k = 0; k < DINNER; ++k) s = fmaf(ys[k], w[k], s);
        os[tid] = s;
    }
    __syncthreads();
    if (tid < 64) {
        const float* w = fc1_w + (size_t)tid * DMODEL;
        float s = fc1_b[tid];
        #pragma unroll 8
        for (int k = 0; k < DMODEL; ++k) s = fmaf(os[k], w[k], s);
        hid[tid] = fmaxf(s, 0.0f);
    }
    __syncthreads();
    if (tid < NCLS) {
        const float* w = fc2_w + (size_t)tid * 64;
        float s = fc2_b[tid];
        #pragma unroll
        for (int k = 0; k < 64; ++k) s = fmaf(hid[k], w[k], s);
        out[b * NCLS + tid] = s;
    }
}

// ---------------------------------------------------------------------------
extern "C" void kernel_launch(void* const* d_in, const int* in_sizes, int n_in,
                              void* d_out, int out_size, void* d_ws, size_t ws_size,
                              hipStream_t stream) {
    (void)in_sizes; (void)n_in; (void)out_size; (void)ws_size;

    const float* x          = (const float*)d_in[0];
    const float* ip_w       = (const float*)d_in[1];
    const float* ip_b       = (const float*)d_in[2];
    const float* in_proj_w  = (const float*)d_in[3];
    const float* conv_w     = (const float*)d_in[4];
    const float* conv_b     = (const float*)d_in[5];
    const float* x_proj_w   = (const float*)d_in[6];
    const float* dt_proj_w  = (const float*)d_in[7];
    const float* dt_proj_b  = (const float*)d_in[8];
    const float* A_log      = (const float*)d_in[9];
    const float* D_skip     = (const float*)d_in[10];
    const float* out_proj_w = (const float*)d_in[11];
    const float* fc1_w      = (const float*)d_in[12];
    const float* fc1_b      = (const float*)d_in[13];
    const float* fc2_w      = (const float*)d_in[14];
    const float* fc2_b      = (const float*)d_in[15];
    float* out = (float*)d_out;

    // workspace carve-up (floats)
    float* ws = (float*)d_ws;
    const size_t n_big = (size_t)NROWS * DINNER;        // 16,384,000
    float* xm      = ws;                 ws += n_big;
    float* delta_b = ws;                 ws += n_big;
    float* du_b    = ws;                 ws += n_big;
    float* Bm_b    = ws;                 ws += (size_t)NROWS * DSTATE;
    float* C_last  = ws;                 ws += BB * DSTATE;
    float* u_last  = ws;                 ws += BB * DINNER;
    float* z_last  = ws;                 ws += BB * DINNER;
    float* y_last  = ws;                 ws += BB * DINNER;

    // 1) input proj + in_proj (xm half) with f32 WMMA
    k_inproj_wmma<<<NROWS / 16, 256, 0, stream>>>(x, ip_w, ip_b, in_proj_w, xm);
    // 2) conv + silu + x_proj (WMMA) + dt_proj + softplus, 16 timesteps/block
    k_conv_proj_wmma<<<BB * TTILES, 256, 0, stream>>>(xm, conv_w, conv_b, x_proj_w,
                                                      dt_proj_w, dt_proj_b,
                                                      delta_b, du_b, Bm_b,
                                                      C_last, u_last);
    // 3) z gate at last timestep
    k_zlast<<<BB, 256, 0, stream>>>(x, ip_w, ip_b, in_proj_w, z_last);
    // 4) selective scan -> y_last
    k_scan<<<(BB * DINNER * DSTATE) / 256, 256, 0, stream>>>(
        delta_b, du_b, Bm_b, A_log, C_last, u_last, D_skip, z_last, y_last);
    // 5) classifier head
    k_head<<<BB, 256, 0, stream>>>(y_last, out_proj_w, fc1_w, fc1_b,
                                   fc2_w, fc2_b, out);
}